// DilateErode_15590731284817
// MI455X (gfx1250) — compile-verified
//
#include <hip/hip_runtime.h>
#include <cstdint>
#include <math.h>

// ---------------------------------------------------------------------------
// Tropical (max-plus / min-minus) "matmul":
//   dilated[b,n] = max_k (x[b,k] + D[k,n]),  eroded[b,n] = min_k (x[b,k] - E[k,n])
// B=1024, F=1024 (+1 zero bias feature), ND=NE=256. out = [eroded | dilated].
// VALU-bound (matrix cores can't do max-plus); CDNA5 path used here is
// async global->LDS staging (ASYNCcnt) + wave32 register tiling + max3/min3.
// ---------------------------------------------------------------------------

#define LDS_AS __attribute__((address_space(3)))
#define GLB_AS __attribute__((address_space(1)))

typedef int v4i __attribute__((ext_vector_type(4)));

#if defined(__has_builtin)
#if __has_builtin(__builtin_amdgcn_global_load_async_to_lds_b128)
#define HAVE_ASYNC_B128_BUILTIN 1
#endif
#if __has_builtin(__builtin_amdgcn_s_wait_asynccnt)
#define HAVE_WAIT_ASYNC_BUILTIN 1
#endif
#endif

struct __align__(16) f4 { float v[4]; };

__device__ __forceinline__ void async_ld_b128(void* lds, const void* g) {
#ifdef HAVE_ASYNC_B128_BUILTIN
  __builtin_amdgcn_global_load_async_to_lds_b128((GLB_AS v4i*)g, (LDS_AS v4i*)lds,
                                                 /*imm offset*/ 0, /*cpol*/ 0);
#else
  uint32_t loff = (uint32_t)(uintptr_t)(LDS_AS void*)lds;
  asm volatile("global_load_async_to_lds_b128 %0, %1, off"
               :: "v"(loff), "v"(g) : "memory");
#endif
}

template <int N>
__device__ __forceinline__ void wait_async() {
#ifdef HAVE_WAIT_ASYNC_BUILTIN
  __builtin_amdgcn_s_wait_asynccnt(N);
#else
  asm volatile("s_wait_asynccnt %0" :: "i"(N));
#endif
}

template <bool IS_MAX>
__device__ __forceinline__ float red2(float a, float b) {
  return IS_MAX ? fmaxf(a, b) : fminf(a, b);
}

// Tile config: block = 128 threads (4 wave32), BM=64 rows x BN=32 cols,
// each thread owns a 4x4 accumulator patch. K staged in chunks of 32 with
// double-buffered async-to-LDS copies.
constexpr int KC  = 32;
constexpr int XST = 36;            // padded x-tile row stride (floats), 16B aligned
constexpr int XS  = 64 * XST;      // one x buffer (floats)
constexpr int WS  = KC * 32;       // one w buffer (floats)

template <bool IS_MAX>
__device__ __forceinline__ void trop_body(const float* __restrict__ x,
                                          const float* __restrict__ W,
                                          float* __restrict__ out,
                                          int out_off,
                                          float* xs, float* ws) {
  const int t  = threadIdx.x;
  const int tx = t & 7;            // output col group: cols cb + tx*4 .. +3
  const int ty = t >> 3;           // output row group: rows rb + ty*4 .. +3
  const int rb = blockIdx.y * 64;
  const int cb = blockIdx.x * 32;

  // staging maps (whole block cooperatively fills the tiles)
  const int xrow = t >> 1, xcol = (t & 1) * 16;  // 64x32 floats, 16 per thread
  const int wrow = t >> 2, wcol = (t & 3) * 8;   // 32x32 floats,  8 per thread

  float acc[4][4];
  const float ini = IS_MAX ? -INFINITY : INFINITY;
#pragma unroll
  for (int i = 0; i < 4; ++i)
#pragma unroll
    for (int j = 0; j < 4; ++j) acc[i][j] = ini;

  auto issue = [&](int kb, int buf) {
    const float* gx = x + (size_t)(rb + xrow) * 1024 + kb + xcol;
    float* lx = xs + buf * XS + xrow * XST + xcol;
    async_ld_b128(lx,      gx);
    async_ld_b128(lx + 4,  gx + 4);
    async_ld_b128(lx + 8,  gx + 8);
    async_ld_b128(lx + 12, gx + 12);
    const float* gw = W + (size_t)(kb + wrow) * 256 + cb + wcol;
    float* lw = ws + buf * WS + wrow * 32 + wcol;
    async_ld_b128(lw,     gw);
    async_ld_b128(lw + 4, gw + 4);
  };

  issue(0, 0);  // preload chunk 0

  for (int c = 0; c < 1024 / KC; ++c) {
    const int buf = c & 1;
    if (c + 1 < 1024 / KC) {
      issue((c + 1) * KC, buf ^ 1);  // prefetch next chunk into other buffer
      wait_async<6>();               // 6 newest (next chunk) may stay in flight
    } else {
      wait_async<0>();
    }
    __syncthreads();                 // everyone's chunk-c data is in LDS

    const float* xb = xs + buf * XS + (ty * 4) * XST;
    const float* wb = ws + buf * WS + tx * 4;

#pragma unroll
    for (int k = 0; k < KC; k += 4) {
      f4 bv[4];
#pragma unroll
      for (int kk = 0; kk < 4; ++kk)
        bv[kk] = *(const f4*)(wb + (k + kk) * 32);     // ds_load_b128
      f4 av[4];
#pragma unroll
      for (int i = 0; i < 4; ++i)
        av[i] = *(const f4*)(xb + i * XST + k);        // ds_load_b128
#pragma unroll
      for (int i = 0; i < 4; ++i) {
#pragma unroll
        for (int j = 0; j < 4; ++j) {
          const float t0 = IS_MAX ? av[i].v[0] + bv[0].v[j] : av[i].v[0] - bv[0].v[j];
          const float t1 = IS_MAX ? av[i].v[1] + bv[1].v[j] : av[i].v[1] - bv[1].v[j];
          const float t2 = IS_MAX ? av[i].v[2] + bv[2].v[j] : av[i].v[2] - bv[2].v[j];
          const float t3 = IS_MAX ? av[i].v[3] + bv[3].v[j] : av[i].v[3] - bv[3].v[j];
          // reassociated so each pair fuses into v_max3/v_min3 with acc
          acc[i][j] = red2<IS_MAX>(red2<IS_MAX>(acc[i][j], t0), t1);
          acc[i][j] = red2<IS_MAX>(red2<IS_MAX>(acc[i][j], t2), t3);
        }
      }
    }
    __syncthreads();                 // buffer safe to overwrite next iteration
  }

  // bias feature: xb[:,1024] == 0 -> contributes +W[1024,n] (max) / -W[1024,n] (min)
  const f4 bb = *(const f4*)(W + (size_t)1024 * 256 + cb + tx * 4);
#pragma unroll
  for (int j = 0; j < 4; ++j) {
    const float tb = IS_MAX ? bb.v[j] : -bb.v[j];
#pragma unroll
    for (int i = 0; i < 4; ++i) acc[i][j] = red2<IS_MAX>(acc[i][j], tb);
  }

#pragma unroll
  for (int i = 0; i < 4; ++i) {
    f4 o;
#pragma unroll
    for (int j = 0; j < 4; ++j) o.v[j] = acc[i][j];
    *(f4*)(out + (size_t)(rb + ty * 4 + i) * 512 + out_off + cb + tx * 4) = o;
  }
}

__global__ __launch_bounds__(128)
void DilateErode_trop_kernel(const float* __restrict__ x,
                             const float* __restrict__ D,
                             const float* __restrict__ E,
                             float* __restrict__ out) {
  __shared__ __align__(16) float xs[2 * XS];
  __shared__ __align__(16) float ws[2 * WS];
  if (blockIdx.z == 0)
    trop_body<false>(x, E, out, /*out_off=*/0,   xs, ws);  // eroded: min(x - E)
  else
    trop_body<true>(x, D, out, /*out_off=*/256, xs, ws);   // dilated: max(x + D)
}

extern "C" void kernel_launch(void* const* d_in, const int* in_sizes, int n_in,
                              void* d_out, int out_size, void* d_ws, size_t ws_size,
                              hipStream_t stream) {
  (void)in_sizes; (void)n_in; (void)d_ws; (void)ws_size; (void)out_size;
  const float* x = (const float*)d_in[0];
  const float* D = (const float*)d_in[1];   // dilations (1025 x 256)
  const float* E = (const float*)d_in[2];   // erosions  (1025 x 256)
  float* out = (float*)d_out;               // (1024 x 512) = [eroded | dilated]

  dim3 grid(256 / 32, 1024 / 64, 2);        // 8 x 16 x 2 = 256 blocks
  dim3 block(128);                          // 4 wave32
  hipLaunchKernelGGL(DilateErode_trop_kernel, grid, block, 0, stream, x, D, E, out);
}